// GCN_34900904247863
// MI455X (gfx1250) — compile-verified
//
#include <hip/hip_runtime.h>
#include <hip/hip_bf16.h>

// GCN 2-layer forward for MI455X (gfx1250), fp32 end-to-end.
// GEMMs use V_WMMA_F32_16X16X4_F32 (wave32, 16x16 tiles) with fused
// epilogues (bias + self-loop init) and fused ReLU-on-load for layer 2.
// Propagation uses edge-parallel global_atomic_add_f32.

typedef float v2f __attribute__((ext_vector_type(2)));
typedef float v8f __attribute__((ext_vector_type(8)));

#define IN_DIM  500
#define HID_DIM 64
#define OUT_DIM 40

// ---------------------------------------------------------------------------
// Degree kernels: deg starts at 1.0 (self loop), +1 per incoming edge, then
// dinv = rsqrt(deg) in place.
// ---------------------------------------------------------------------------
__global__ void gcn_deg_init(float* __restrict__ deg, int n) {
    int i = blockIdx.x * blockDim.x + threadIdx.x;
    if (i < n) deg[i] = 1.0f;
}

__global__ void gcn_deg_count(const long long* __restrict__ dst,
                              float* __restrict__ deg, int n_edges) {
    int e = blockIdx.x * blockDim.x + threadIdx.x;
    if (e < n_edges) atomicAdd(&deg[(int)dst[e]], 1.0f);
}

__global__ void gcn_deg_rsqrt(float* __restrict__ deg, int n) {
    int i = blockIdx.x * blockDim.x + threadIdx.x;
    if (i < n) deg[i] = rsqrtf(deg[i]);
}

// ---------------------------------------------------------------------------
// Fused WMMA GEMM:
//   H[n_rows x NOUT]   = act(A)[n_rows x K] * B[K x NOUT]
//   G[n_rows x NOUT]   = H * dinv[row]^2 + bias[col]   (self-loop + bias init)
// act = ReLU when RELU_A (applied to A on load; A itself is not modified).
// One wave computes a 16-row x (NT*16)-col stripe; n_rows % 16 == 0 holds
// (100000). NOUT may be < NT*16 (column-masked loads/stores).
//
// VGPR layouts per CDNA5 ISA 7.12.2 (32-bit, wave32):
//   A 16x4 : lanes 0-15 row M=lane, VGPR{0,1} = K{0,1}; lanes 16-31 K{2,3}
//   B 4x16 : VGPR0: lanes0-15 (K+0, N=lane), lanes16-31 (K+2, N=lane-16)
//            VGPR1: lanes0-15 (K+1, N),      lanes16-31 (K+3, N)
//   C 16x16: VGPR i: lanes0-15 row M=i, lanes16-31 row M=i+8
// ---------------------------------------------------------------------------
template <int K, int NOUT, int NT, bool RELU_A>
__global__ __launch_bounds__(256) void gcn_gemm_fused(
    const float* __restrict__ A, const float* __restrict__ B,
    const float* __restrict__ dinv, const float* __restrict__ bias,
    float* __restrict__ H, float* __restrict__ G, int n_rows)
{
    const int lane = threadIdx.x & 31;
    const int wave = blockIdx.x * (blockDim.x >> 5) + (threadIdx.x >> 5);
    const int m0   = wave << 4;
    if (m0 >= n_rows) return;              // wave-uniform exit: EXEC stays full

    const int half = lane >> 4;            // 0: lanes 0-15, 1: lanes 16-31
    const int l16  = lane & 15;
    const float* __restrict__ arow = A + (size_t)(m0 + l16) * K;

    v8f acc[NT];
    #pragma unroll
    for (int t = 0; t < NT; ++t) acc[t] = (v8f){};

    for (int k = 0; k < K; k += 4) {
        // stream-prefetch the A row ahead of use (global_prefetch_b8)
        if (k + 64 < K) __builtin_prefetch(arow + k + 64, 0, 1);

        const int kk = k + 2 * half;       // this half-wave's K pair
        // A is streamed exactly once per kernel: non-temporal load (TH_NT)
        v2f a = __builtin_nontemporal_load((const v2f*)(arow + kk));
        if (RELU_A) { a.x = fmaxf(a.x, 0.0f); a.y = fmaxf(a.y, 0.0f); }

        #pragma unroll
        for (int t = 0; t < NT; ++t) {
            const int n = t * 16 + l16;
            v2f b;
            b.x = (n < NOUT) ? B[(kk + 0) * NOUT + n] : 0.0f;
            b.y = (n < NOUT) ? B[(kk + 1) * NOUT + n] : 0.0f;
            acc[t] = __builtin_amdgcn_wmma_f32_16x16x4_f32(
                /*neg_a=*/false, a, /*neg_b=*/false, b,
                /*c_mod=*/(short)0, acc[t],
                /*reuse_a=*/false, /*reuse_b=*/false);
        }
    }

    // Fused epilogue: H = acc;  G = acc*dinv^2 + bias
    #pragma unroll
    for (int i = 0; i < 8; ++i) {
        const int r  = m0 + i + 8 * half;  // output row for this VGPR index
        const float di  = dinv[r];
        const float dsq = di * di;
        #pragma unroll
        for (int t = 0; t < NT; ++t) {
            const int n = t * 16 + l16;
            if (n < NOUT) {
                const float v = acc[t][i];
                const size_t o = (size_t)r * NOUT + n;
                H[o] = v;
                G[o] = v * dsq + bias[n];
            }
        }
    }
}

// ---------------------------------------------------------------------------
// out[dst,f] += h[src,f] * dinv[src]*dinv[dst]   edge-major, feature-minor
// ---------------------------------------------------------------------------
template <int D>
__global__ void gcn_edge_scatter(const long long* __restrict__ src,
                                 const long long* __restrict__ dst,
                                 const float* __restrict__ h,
                                 const float* __restrict__ dinv,
                                 float* __restrict__ out, int n_edges)
{
    int idx = blockIdx.x * blockDim.x + threadIdx.x;
    int e = idx / D;
    int f = idx - e * D;
    if (e >= n_edges) return;
    int s = (int)src[e];
    int d = (int)dst[e];
    float w = dinv[s] * dinv[d];
    atomicAdd(out + (size_t)d * D + f, h[(size_t)s * D + f] * w);
}

// ---------------------------------------------------------------------------
extern "C" void kernel_launch(void* const* d_in, const int* in_sizes, int n_in,
                              void* d_out, int out_size, void* d_ws, size_t ws_size,
                              hipStream_t stream)
{
    const float*     x   = (const float*)d_in[0];      // [N, 500]
    const float*     W1  = (const float*)d_in[1];      // [500, 64]
    const float*     b1  = (const float*)d_in[2];      // [64]
    const float*     W2  = (const float*)d_in[3];      // [64, 40]
    const float*     b2  = (const float*)d_in[4];      // [40]
    const long long* ei  = (const long long*)d_in[5];  // [2, E] int64

    const int n_nodes = in_sizes[0] / IN_DIM;          // 100000
    const int n_edges = in_sizes[5] / 2;               // 1600000
    const long long* src = ei;
    const long long* dst = ei + n_edges;

    // workspace: dinv | hbuf (N*64, reused as N*40) | abuf (N*64)
    float* dinv = (float*)d_ws;
    float* hbuf = dinv + n_nodes;                      // raw GEMM outputs
    float* abuf = hbuf + (size_t)n_nodes * HID_DIM;    // aggregation buffer
    float* outf = (float*)d_out;                       // [N, 40]

    const int T = 256;
    auto blocks = [](long long total, int t) { return (int)((total + t - 1) / t); };

    // --- degrees -> dinv ---
    gcn_deg_init <<<blocks(n_nodes, T), T, 0, stream>>>(dinv, n_nodes);
    gcn_deg_count<<<blocks(n_edges, T), T, 0, stream>>>(dst, dinv, n_edges);
    gcn_deg_rsqrt<<<blocks(n_nodes, T), T, 0, stream>>>(dinv, n_nodes);

    const int waves = (n_nodes + 15) / 16;             // 6250 (16 rows/wave)
    const int gblk  = blocks(waves, T / 32);           // 8 waves / block

    // --- layer 1: hbuf = x@W1 ; abuf = hbuf*dinv^2 + b1 (fused) ---
    gcn_gemm_fused<IN_DIM, HID_DIM, 4, false><<<gblk, T, 0, stream>>>(
        x, W1, dinv, b1, hbuf, abuf, n_nodes);

    // --- propagate layer 1: abuf[d] += hbuf[s]*dinv[s]*dinv[d] ---
    gcn_edge_scatter<HID_DIM><<<blocks((long long)n_edges * HID_DIM, T), T, 0, stream>>>(
        src, dst, hbuf, dinv, abuf, n_edges);

    // --- layer 2: hbuf = relu(abuf)@W2 ; d_out = hbuf*dinv^2 + b2 (fused) ---
    gcn_gemm_fused<HID_DIM, OUT_DIM, 3, true><<<gblk, T, 0, stream>>>(
        abuf, W2, dinv, b2, hbuf, outf, n_nodes);

    // --- propagate layer 2 into d_out ---
    gcn_edge_scatter<OUT_DIM><<<blocks((long long)n_edges * OUT_DIM, T), T, 0, stream>>>(
        src, dst, hbuf, dinv, outf, n_edges);
}